// Sinkhorn_50921132261709
// MI455X (gfx1250) — compile-verified
//
#include <hip/hip_runtime.h>
#include <hip/hip_bf16.h>
#include <stdint.h>

// Sinkhorn (B=64, R=1024, C=1024, TAU=1, 10 iters) via row/col potentials:
//   row pass: a[r] = LSE_{c<nc}(s[r][c] - b[c])
//   col pass: b[c] = LSE_{r<nr}(s[r][c] - a[r])   (TDM tiles -> LDS, online LSE)
//   final   : out  = valid ? exp(s - a - b) : 0
// Memory-bound: ~1.7 GB total traffic -> ~75us at 23.3 TB/s. No GEMM structure,
// so WMMA does not apply; the CDNA5 path used is the Tensor Data Mover + tensorcnt.

#define BATCH 64
#define RDIM  1024
#define CDIM  1024
#define NEGF  (-1e30f)
#define TR    16   // rows per TDM tile in col pass
#define CB    256  // columns per block in col pass

typedef unsigned int u32x4 __attribute__((ext_vector_type(4)));
typedef int          i32x8 __attribute__((ext_vector_type(8)));
typedef int          i32x4 __attribute__((ext_vector_type(4)));

#if defined(__HIP_DEVICE_COMPILE__) && __has_builtin(__builtin_amdgcn_tensor_load_to_lds)
#define HAVE_TDM 1
#else
#define HAVE_TDM 0
#endif

#if defined(__HIP_DEVICE_COMPILE__)
#if HAVE_TDM
#warning "CDNA5 TDM path ACTIVE: __builtin_amdgcn_tensor_load_to_lds"
#else
#warning "TDM builtin NOT available: cooperative-copy fallback in col pass"
#endif
#endif

#if HAVE_TDM
// Build a D# (CDNA5 ISA) for a 2D tile: TR rows x CB cols of f32,
// row stride CDIM elements, and issue TENSOR_LOAD_TO_LDS. Per-wave op (EXEC ignored);
// caller must guard so only one wave issues it.
__device__ __forceinline__ void tdm_load_tile(const float* gsrc, void* lds_dst) {
  unsigned lds = (unsigned)(unsigned long long)(size_t)lds_dst; // low 32 bits = LDS byte offset
  unsigned long long ga = (unsigned long long)(size_t)gsrc;
  u32x4 g0;
  g0.x = 1u;                                        // count=1 (valid descriptor)
  g0.y = lds;                                       // lds_addr
  g0.z = (unsigned)ga;                              // global_addr[31:0]
  g0.w = (unsigned)((ga >> 32) & 0x1FFFFFFu)        // global_addr[56:32]
         | (2u << 30);                              // type=2 ("image")
  i32x8 g1;
  g1[0] = (int)(2u << 16);             // data_size=2 (4 bytes); wg_mask=0 (no cluster)
  g1[1] = (int)((unsigned)CB << 16);   // tensor_dim0 = CB (low 16 bits at [63:48])
  g1[2] = (int)((unsigned)TR << 16);   // tensor_dim0 hi=0 ; tensor_dim1 = TR
  g1[3] = (int)((unsigned)CB << 16);   // tensor_dim1 hi=0 ; tile_dim0 = CB
  g1[4] = TR;                          // tile_dim1 = TR ; tile_dim2 = 0
  g1[5] = CDIM;                        // tensor_dim0_stride (low 32)
  g1[6] = 0;
  g1[7] = 0;
  i32x4 z4 = {0, 0, 0, 0};
#if __clang_major__ >= 23
  i32x8 z8 = {0, 0, 0, 0, 0, 0, 0, 0};
  __builtin_amdgcn_tensor_load_to_lds(g0, g1, z4, z4, z8, 0);
#else
  __builtin_amdgcn_tensor_load_to_lds(g0, g1, z4, z4, 0);
#endif
}
#endif

__global__ __launch_bounds__(256) void sinkhorn_zero(float* __restrict__ p, int n) {
  int i = blockIdx.x * 256 + threadIdx.x;
  if (i < n) p[i] = 0.f;
}

// Row pass: one wave per row; lanes cover 128-col chunks as float4; online LSE
// per lane, then 5-step shfl_xor merge of (max, sum). Rows >= nrows skipped.
__global__ __launch_bounds__(256) void sinkhorn_rowpass(
    const float* __restrict__ s, const float* __restrict__ bpot,
    float* __restrict__ apot, const int* __restrict__ nrows,
    const int* __restrict__ ncols) {
  const int lane = threadIdx.x & 31;
  const int rowg = blockIdx.x * 8 + (threadIdx.x >> 5);
  const int bb = rowg >> 10;
  const int r  = rowg & (RDIM - 1);
  if (r >= nrows[bb]) return;                  // never read downstream
  const int nc = ncols[bb];
  const float* srow = s + (((size_t)bb << 20) + ((size_t)r << 10));
  const float* brow = bpot + (bb << 10);

  float m = NEGF, acc = 0.f;
  const int kmax = (nc + 127) >> 7;            // 128-col chunks actually needed
  for (int k = 0; k < kmax; ++k) {
    const int j0 = (k << 7) + (lane << 2);
    const float4 sv = *(const float4*)(srow + j0);
    const float4 bv = *(const float4*)(brow + j0);
    const float x0 = sv.x - bv.x, x1 = sv.y - bv.y;
    const float x2 = sv.z - bv.z, x3 = sv.w - bv.w;
    const bool v0 = (j0 + 0) < nc, v1 = (j0 + 1) < nc;
    const bool v2 = (j0 + 2) < nc, v3 = (j0 + 3) < nc;
    const float cm = fmaxf(fmaxf(v0 ? x0 : NEGF, v1 ? x1 : NEGF),
                           fmaxf(v2 ? x2 : NEGF, v3 ? x3 : NEGF));
    const float mN = fmaxf(m, cm);
    float e = acc * __expf(m - mN);
    e += v0 ? __expf(x0 - mN) : 0.f;
    e += v1 ? __expf(x1 - mN) : 0.f;
    e += v2 ? __expf(x2 - mN) : 0.f;
    e += v3 ? __expf(x3 - mN) : 0.f;
    acc = e; m = mN;
  }
  // cross-lane (max, sum-of-exp) merge
  for (int off = 16; off > 0; off >>= 1) {
    const float mo = __shfl_xor(m, off, 32);
    const float so = __shfl_xor(acc, off, 32);
    const float mN = fmaxf(m, mo);
    acc = acc * __expf(m - mN) + so * __expf(mo - mN);
    m = mN;
  }
  if (lane == 0) apot[(bb << 10) + r] = m + __logf(acc);
}

// Col pass: block = 256 threads = 256 columns of one batch; reduce over rows
// (< nrows) with an online LSE merging 4 rows per step (shorter dependency
// chain, 1.25 exp/elem). Tiles of TRxCB f32 are DMA'd into LDS by the Tensor
// Data Mover, double-buffered with s_wait_tensorcnt + workgroup barriers.
__global__ __launch_bounds__(256) void sinkhorn_colpass(
    const float* __restrict__ s, const float* __restrict__ apot,
    float* __restrict__ bpot, const int* __restrict__ nrows,
    const int* __restrict__ ncols) {
  __shared__ float tile[2][TR * CB];
  __shared__ float arow[2][TR];
  const int tid = threadIdx.x;
  const int bb = blockIdx.x >> 2;
  const int c0 = (blockIdx.x & 3) << 8;
  const int c  = c0 + tid;
  const int nr = nrows[bb];
  const int nc = ncols[bb];
  const int ntiles = (nr + TR - 1) / TR;       // nr >= 128 -> >= 8 tiles
  const float* sbase = s + (((size_t)bb << 20) + (size_t)c0);
  const float* abase = apot + (bb << 10);

  // prologue: tile 0 into buffer 0
  if (tid < TR) arow[0][tid] = abase[tid];
#if HAVE_TDM
  if (tid == 0) {
    tdm_load_tile(sbase, &tile[0][0]);
    __builtin_amdgcn_s_wait_tensorcnt(0);
  }
#else
  for (int rr = 0; rr < TR; ++rr) tile[0][rr * CB + tid] = sbase[(size_t)rr * CDIM + tid];
#endif
  __syncthreads();

  float m = NEGF, acc = 0.f;
  for (int t = 0; t < ntiles; ++t) {
    const int buf = t & 1;
    const int nxt = buf ^ 1;
    if (t + 1 < ntiles) {
      if (tid < TR) arow[nxt][tid] = abase[(t + 1) * TR + tid];
#if HAVE_TDM
      if (tid == 0) tdm_load_tile(sbase + (size_t)(t + 1) * TR * CDIM, &tile[nxt][0]);
#else
      for (int rr = 0; rr < TR; ++rr)
        tile[nxt][rr * CB + tid] = sbase[((size_t)(t + 1) * TR + rr) * CDIM + tid];
#endif
    }
    int rmax = nr - t * TR;
    if (rmax > TR) rmax = TR;
    const float* tp = &tile[buf][tid];
    const float* ap = &arow[buf][0];
    int rr = 0;
    #pragma unroll
    for (; rr + 4 <= TR; rr += 4) {            // full 4-row groups
      if (rr + 4 > rmax) break;
      const float x0 = tp[(rr + 0) * CB] - ap[rr + 0];
      const float x1 = tp[(rr + 1) * CB] - ap[rr + 1];
      const float x2 = tp[(rr + 2) * CB] - ap[rr + 2];
      const float x3 = tp[(rr + 3) * CB] - ap[rr + 3];
      const float cm = fmaxf(fmaxf(x0, x1), fmaxf(x2, x3));
      const float mN = fmaxf(m, cm);
      acc = acc * __expf(m - mN) + __expf(x0 - mN) + __expf(x1 - mN) +
            __expf(x2 - mN) + __expf(x3 - mN);
      m = mN;
    }
    for (; rr < rmax; ++rr) {                  // remainder rows (last tile)
      const float x = tp[rr * CB] - ap[rr];
      const float mN = fmaxf(m, x);
      acc = acc * __expf(m - mN) + __expf(x - mN);
      m = mN;
    }
#if HAVE_TDM
    if (t + 1 < ntiles) {
      if (tid == 0) __builtin_amdgcn_s_wait_tensorcnt(0);
    }
#endif
    __syncthreads();
  }
  if (c < nc) bpot[(bb << 10) + c] = m + __logf(acc);
}

// Finalize: out = valid ? exp(s - a - b) : 0, float4 per thread, one block per row.
__global__ __launch_bounds__(256) void sinkhorn_finalize(
    const float* __restrict__ s, const float* __restrict__ apot,
    const float* __restrict__ bpot, const int* __restrict__ nrows,
    const int* __restrict__ ncols, float* __restrict__ out) {
  const int tid = threadIdx.x;
  const int bb = blockIdx.x >> 10;
  const int r  = blockIdx.x & (RDIM - 1);
  const size_t base = ((size_t)blockIdx.x << 10);
  const int c0 = tid << 2;
  float4 o = make_float4(0.f, 0.f, 0.f, 0.f);
  if (r < nrows[bb]) {
    const int nc = ncols[bb];
    const float ar = apot[(bb << 10) + r];
    const float4 sv = *(const float4*)(s + base + c0);
    const float4 bv = *(const float4*)(bpot + (bb << 10) + c0);
    o.x = (c0 + 0 < nc) ? __expf(sv.x - ar - bv.x) : 0.f;
    o.y = (c0 + 1 < nc) ? __expf(sv.y - ar - bv.y) : 0.f;
    o.z = (c0 + 2 < nc) ? __expf(sv.z - ar - bv.z) : 0.f;
    o.w = (c0 + 3 < nc) ? __expf(sv.w - ar - bv.w) : 0.f;
  }
  *(float4*)(out + base + c0) = o;
}

extern "C" void kernel_launch(void* const* d_in, const int* in_sizes, int n_in,
                              void* d_out, int out_size, void* d_ws, size_t ws_size,
                              hipStream_t stream) {
  const float* s     = (const float*)d_in[0];
  const int*   nrows = (const int*)d_in[1];
  const int*   ncols = (const int*)d_in[2];
  float* out  = (float*)d_out;
  float* apot = (float*)d_ws;                 // [BATCH][RDIM]
  float* bpot = apot + BATCH * RDIM;          // [BATCH][CDIM]

  const int npot = BATCH * (RDIM + CDIM);
  sinkhorn_zero<<<(npot + 255) / 256, 256, 0, stream>>>(apot, npot);

  for (int it = 0; it < 10; ++it) {
    if ((it & 1) == 0)
      sinkhorn_rowpass<<<BATCH * RDIM / 8, 256, 0, stream>>>(s, bpot, apot, nrows, ncols);
    else
      sinkhorn_colpass<<<BATCH * (CDIM / CB), 256, 0, stream>>>(s, apot, bpot, nrows, ncols);
  }

  sinkhorn_finalize<<<BATCH * RDIM, 256, 0, stream>>>(s, apot, bpot, nrows, ncols, out);
}